// RelativisticDeerskinLayer_979252544419
// MI455X (gfx1250) — compile-verified
//
#include <hip/hip_runtime.h>
#include <math.h>

typedef __attribute__((ext_vector_type(2))) float v2f;
typedef __attribute__((ext_vector_type(8))) float v8f;

#define N_NEUR 512
#define DDIM   32
#define TSTEPS 8192
#define OMEGA_C  10.0f
#define ALPHA_C  0.5f
#define DT_C     1e-3f
#define TWO_PI_C 6.2831853071795864769f

// constant-index 8-way select from a WMMA f32 accumulator (avoids scratch)
__device__ __forceinline__ float sel8(const v8f v, int r) {
    return (r & 4) ? ((r & 2) ? ((r & 1) ? v[7] : v[6]) : ((r & 1) ? v[5] : v[4]))
                   : ((r & 2) ? ((r & 1) ? v[3] : v[2]) : ((r & 1) ? v[1] : v[0]));
}

__device__ __forceinline__ float butterfly_sum32(float s) {
    s += __shfl_xor(s, 16);
    s += __shfl_xor(s, 8);
    s += __shfl_xor(s, 4);
    s += __shfl_xor(s, 2);
    s += __shfl_xor(s, 1);
    return s;   // commutative pairing -> bit-identical in every lane
}

__global__ __launch_bounds__(1024)
void deerskin_persistent_kernel(const float* __restrict__ Xr,
                                const float* __restrict__ Xi,
                                const float* __restrict__ tw,
                                const float* __restrict__ W0r,
                                const float* __restrict__ W0i,
                                const float* __restrict__ phi0,
                                const float* __restrict__ beta0,
                                float* __restrict__ out) {
    __shared__ float partials[2][32];

    const int tid  = threadIdx.x;
    const int lane = tid & 31;
    const int wid  = tid >> 5;           // 32 waves, 16 neurons each
    const int hi   = (lane >> 4) & 1;    // lane half (selects K pair in A/B layout)
    const int m    = lane & 15;          // neuron within tile (A-layout row)
    const int neuron = wid * 16 + m;     // replicated on lanes l and l+16

    // shuffle source that maps C/D-layout replication to per-lane neuron l%16
    const int srcLane = (lane & 15) | ((lane & 8) << 1);

    // ---- persistent state: W rows in WMMA A-layout registers ----
    v2f Wr[8], Wi[8];
#pragma unroll
    for (int c = 0; c < 8; ++c) {
        const int d = c * 4 + hi * 2;
        Wr[c][0] = W0r[neuron * DDIM + d];
        Wr[c][1] = W0r[neuron * DDIM + d + 1];
        Wi[c][0] = W0i[neuron * DDIM + d];
        Wi[c][1] = W0i[neuron * DDIM + d + 1];
    }
    float phi  = phi0[neuron];
    float beta = beta0[neuron];
    float lt   = 0.0f;
    float err  = 0.0f;   // replicated scalar state, kept coherent across all threads

    float* outs   = out;                               // [T]
    float* betas  = out + TSTEPS;                      // [T,N]
    float* gammas = out + TSTEPS + (size_t)TSTEPS * N_NEUR;  // [T,N]

    for (int t = 0; t < TSTEPS; ++t) {
        if (tid == 0 && t + 8 < TSTEPS) {              // TH=NT-ish stream-ahead
            __builtin_prefetch(Xr + (size_t)(t + 8) * DDIM, 0, 1);
            __builtin_prefetch(Xi + (size_t)(t + 8) * DDIM, 0, 1);
        }
        const float tgt = tw[t];
        const float* xrow_r = Xr + (size_t)t * DDIM;
        const float* xrow_i = Xi + (size_t)t * DDIM;

        // x chunks in WMMA B-layout (value replicated across all 16 B columns)
        v2f xr[8], xi[8];
#pragma unroll
        for (int c = 0; c < 8; ++c) {
            const int d = c * 4 + hi * 2;
            xr[c][0] = xrow_r[d]; xr[c][1] = xrow_r[d + 1];
            xi[c][0] = xrow_i[d]; xi[c][1] = xrow_i[d + 1];
        }

        // time dilation (old beta), local time, oscillator phase
        const float gamma = __expf(-ALPHA_C * beta);
        const float dtl   = gamma * DT_C;
        lt += dtl;
        const float theta = OMEGA_C * lt + phi;
        const float costh = __cosf(theta);
        const float sinth = __sinf(theta);

        // ---- Z = W @ x : 32x v_wmma_f32_16x16x4_f32, two indep chains ----
        v8f accA = {};   // sum Wr*xr
        v8f accB = {};   // sum Wi*xi   (Zr = accA - accB)
        v8f accI = {};   // sum Wr*xi + Wi*xr
#pragma unroll
        for (int c = 0; c < 8; ++c) {
            accA = __builtin_amdgcn_wmma_f32_16x16x4_f32(false, Wr[c], false, xr[c], (short)0, accA, false, false);
            accI = __builtin_amdgcn_wmma_f32_16x16x4_f32(false, Wr[c], false, xi[c], (short)0, accI, false, false);
        }
#pragma unroll
        for (int c = 0; c < 8; ++c) {
            accB = __builtin_amdgcn_wmma_f32_16x16x4_f32(false, Wi[c], false, xi[c], (short)0, accB, false, false);
            accI = __builtin_amdgcn_wmma_f32_16x16x4_f32(false, Wi[c], false, xr[c], (short)0, accI, false, false);
        }

        // extract this lane's neuron Z from the replicated C/D layout
        const int r = lane & 7;
        float zrL = sel8(accA, r) - sel8(accB, r);
        float ziL = sel8(accI, r);
        const float zr = __shfl(zrL, srcLane);
        const float zi = __shfl(ziL, srcLane);

        // |Z| * relu(cos(theta - angle(Z))) == relu(Zr*cos + Zi*sin)
        const float Y = fmaxf(zr * costh + zi * sinth, 0.0f);

        // ---- out = sum_n Y*cos(theta): wave partial then cross-wave ----
        const int par = t & 1;
        float p = (hi == 0) ? Y * costh : 0.0f;   // count each neuron once
        p = butterfly_sum32(p);
        if (lane == 0) partials[par][wid] = p;
        __syncthreads();
        float outv = butterfly_sum32(partials[par][lane]);

        // ---- thermodynamic feedback (replicated scalar path) ----
        const float raw  = fabsf(tgt - outv);
        err = 0.99f * err + 0.01f * raw;
        const float rel  = err / (fabsf(tgt) + 0.01f);
        const float btar = 3.5f * __expf(-5.0f * rel);
        const float tau  = (btar > beta) ? (0.5f / (gamma + 1e-6f)) : 0.03f;
        const float a_s  = 1.0f - __expf(-DT_C / tau);
        float bnew = beta + a_s * (btar - beta);
        bnew = fminf(fmaxf(bnew, 0.005f), 5.0f);

        // ---- Hebbian growth / viscous decay on A-layout W registers ----
        const float gY   = 0.05f * Y;
        const float cdec = bnew * Y * Y;
#pragma unroll
        for (int c = 0; c < 8; ++c) {
#pragma unroll
            for (int j = 0; j < 2; ++j) {
                const float wr = Wr[c][j], wi = Wi[c][j];
                Wr[c][j] = wr + (gY * xr[c][j] - cdec * wr) * dtl;
                Wi[c][j] = wi + (gY * xi[c][j] - cdec * wi) * dtl;
            }
        }

        // phase pull + wrap to [0, 2pi)
        phi = phi + (2.0f / (float)N_NEUR) * (-tgt * sinth) * dtl;
        phi = phi - TWO_PI_C * floorf(phi / TWO_PI_C);

        beta = bnew;

        // ---- trace outputs ----
        if (tid == 0) outs[t] = outv;
        if (hi == 0) {
            betas[(size_t)t * N_NEUR + neuron]  = bnew;
            gammas[(size_t)t * N_NEUR + neuron] = gamma;
        }
        // no second barrier: partials double-buffered, 2-barrier separation holds
    }
}

extern "C" void kernel_launch(void* const* d_in, const int* in_sizes, int n_in,
                              void* d_out, int out_size, void* d_ws, size_t ws_size,
                              hipStream_t stream) {
    const float* Xr    = (const float*)d_in[0];
    const float* Xi    = (const float*)d_in[1];
    const float* tw    = (const float*)d_in[2];
    const float* W0r   = (const float*)d_in[3];
    const float* W0i   = (const float*)d_in[4];
    const float* phi0  = (const float*)d_in[5];
    const float* beta0 = (const float*)d_in[6];
    float* out = (float*)d_out;

    deerskin_persistent_kernel<<<dim3(1), dim3(1024), 0, stream>>>(
        Xr, Xi, tw, W0r, W0i, phi0, beta0, out);
}